// GCNWrapper_70987219469122
// MI455X (gfx1250) — compile-verified
//
#include <hip/hip_runtime.h>

typedef __attribute__((ext_vector_type(2))) float v2f;
typedef __attribute__((ext_vector_type(8))) float v8f;
typedef __attribute__((ext_vector_type(4))) int   v4i;

#define N_NODES 40000
#define N_EDGES 640000
#define IN_DIM  256
#define OUT_DIM 128
// LDS row stride for staged W: 128 + 8 pad floats. Rows k and k+2 (read by the
// two half-waves of a WMMA B fragment) are offset 2*136 = 272 ≡ 16 (mod 64)
// banks apart -> the two 16-lane groups hit disjoint bank groups, no conflicts.
#define LDS_STRIDE 136
#define LDS_BYTES  (IN_DIM * LDS_STRIDE * 4)   // 139264 B (< 320 KB/WGP)

#ifndef USE_ASYNC_LDS
#define USE_ASYNC_LDS 1
#endif

// ---------------------------------------------------------------------------
// 1) deg[i] = 1.0 (self-loop contribution baked in)
__global__ void k_init_deg(float* __restrict__ deg) {
    int i = blockIdx.x * blockDim.x + threadIdx.x;
    if (i < N_NODES) deg[i] = 1.0f;
}

// 2) deg[dst[e]] += 1 for every edge
__global__ void k_deg_atomic(const long long* __restrict__ dst,
                             float* __restrict__ deg) {
    int e = blockIdx.x * blockDim.x + threadIdx.x;
    if (e < N_EDGES) atomicAdd(&deg[(int)dst[e]], 1.0f);
}

// 3) dinv = rsqrt(deg)  (deg >= 1 always, so no zero guard needed)
__global__ void k_dinv(float* __restrict__ deg) {
    int i = blockIdx.x * blockDim.x + threadIdx.x;
    if (i < N_NODES) deg[i] = rsqrtf(deg[i]);
}

// ---------------------------------------------------------------------------
// 4) xw = x @ W using V_WMMA_F32_16X16X4_F32 (f32 in/out — matches reference).
//    W (128 KB) is staged once per block into LDS (CDNA5: 320 KB per WGP)
//    via the async global->LDS path (GLOBAL_LOAD_ASYNC_TO_LDS_B128, tracked
//    by ASYNCcnt), cutting L2 W-traffic from ~320 MB (per-wave re-reads) to
//    ~43 MB (per-block). One wave owns a 16-row M tile and computes all 8
//    N tiles, reusing each A fragment across 8 WMMAs. K loop: 256/4 = 64.
__global__ void k_gemm_wmma(const float* __restrict__ x,
                            const float* __restrict__ W,
                            float* __restrict__ xw) {
    extern __shared__ float sW[];

    // --- cooperative stage of W into LDS (all 256 threads, 32 float4 each) ---
    {
        const int t = threadIdx.x;
#pragma unroll
        for (int i = 0; i < 32; ++i) {
            const int idx = t + 256 * i;        // float4 chunk id, 8192 total
            const int kk  = idx >> 5;           // row k   = idx / 32
            const int n4  = idx & 31;           // col/4   = idx % 32
#if USE_ASYNC_LDS && __has_builtin(__builtin_amdgcn_global_load_async_to_lds_b128)
            __builtin_amdgcn_global_load_async_to_lds_b128(
                (__attribute__((address_space(1))) v4i*)(W + 4 * (size_t)idx),
                (__attribute__((address_space(3))) v4i*)(&sW[kk * LDS_STRIDE + n4 * 4]),
                0, 0);
#else
            float4 v = ((const float4*)W)[idx];
            *(float4*)&sW[kk * LDS_STRIDE + n4 * 4] = v;
#endif
        }
#if USE_ASYNC_LDS && __has_builtin(__builtin_amdgcn_global_load_async_to_lds_b128)
#if __has_builtin(__builtin_amdgcn_s_wait_asynccnt)
        __builtin_amdgcn_s_wait_asynccnt(0);
#else
        asm volatile("s_wait_asynccnt 0x0" ::: "memory");
#endif
#endif
        __syncthreads();   // all waves of the block participate
    }

    const int wave = (int)((blockIdx.x * blockDim.x + threadIdx.x) >> 5);
    const int lane = threadIdx.x & 31;
    if (wave >= N_NODES / 16) return;      // wave-uniform: EXEC stays all-1s

    const int m0   = wave * 16;
    const int half = lane >> 4;            // 0: K+0/1   1: K+2/3
    const int l16  = lane & 15;
    const int arow = m0 + l16;

    v8f acc[8];
#pragma unroll
    for (int t = 0; t < 8; ++t) acc[t] = (v8f)(0.0f);

    const float* xr = x + (size_t)arow * IN_DIM;

    for (int k = 0; k < IN_DIM; k += 4) {
        const int ka = k + half * 2;
        v2f a;
        a.x = xr[ka];
        a.y = xr[ka + 1];
        const float* wrow0 = &sW[(size_t)ka * LDS_STRIDE + l16];  // row ka
        const float* wrow1 = wrow0 + LDS_STRIDE;                  // row ka+1
#pragma unroll
        for (int t = 0; t < 8; ++t) {
            v2f b;
            b.x = wrow0[t * 16];
            b.y = wrow1[t * 16];
            acc[t] = __builtin_amdgcn_wmma_f32_16x16x4_f32(
                false, a, false, b, (short)0, acc[t], false, false);
        }
    }

    const int crow = m0 + half * 8;
    const int ccol = l16;
#pragma unroll
    for (int t = 0; t < 8; ++t) {
        const int n0 = t * 16;
#pragma unroll
        for (int r = 0; r < 8; ++r) {
            xw[(size_t)(crow + r) * OUT_DIM + n0 + ccol] = acc[t][r];
        }
    }
}

// ---------------------------------------------------------------------------
// 5) out = dinv[i]^2 * xw[i] + b   (self-loop term + bias, before edge atomics)
__global__ void k_out_init(const float* __restrict__ dinv,
                           const float* __restrict__ xw,
                           const float* __restrict__ b,
                           float* __restrict__ out) {
    int idx = blockIdx.x * blockDim.x + threadIdx.x;
    if (idx < N_NODES * OUT_DIM) {
        int i = idx >> 7;          // / OUT_DIM
        int c = idx & (OUT_DIM - 1);
        float di = dinv[i];
        out[idx] = di * di * xw[idx] + b[c];
    }
}

// 6) one wave per edge: out[dst] += dinv[src]*dinv[dst] * xw[src]
//    lane L handles channels 4L..4L+3 (coalesced float4 gather, 4 f32 atomics)
__global__ void k_scatter(const long long* __restrict__ src,
                          const long long* __restrict__ dst,
                          const float* __restrict__ dinv,
                          const float* __restrict__ xw,
                          float* __restrict__ out) {
    int wave = (int)((blockIdx.x * blockDim.x + threadIdx.x) >> 5);
    int lane = threadIdx.x & 31;
    if (wave >= N_EDGES) return;

    const int s = (int)src[wave];
    const int d = (int)dst[wave];
    const float nrm = dinv[s] * dinv[d];

    const float4* xs = (const float4*)(xw + (size_t)s * OUT_DIM);
    float4 v = xs[lane];

    float* o = out + (size_t)d * OUT_DIM + lane * 4;
    atomicAdd(o + 0, nrm * v.x);
    atomicAdd(o + 1, nrm * v.y);
    atomicAdd(o + 2, nrm * v.z);
    atomicAdd(o + 3, nrm * v.w);
}

// 7) out = relu(out)
__global__ void k_relu(float* __restrict__ out) {
    int idx = blockIdx.x * blockDim.x + threadIdx.x;
    if (idx < N_NODES * OUT_DIM) out[idx] = fmaxf(out[idx], 0.0f);
}

// ---------------------------------------------------------------------------
extern "C" void kernel_launch(void* const* d_in, const int* in_sizes, int n_in,
                              void* d_out, int out_size, void* d_ws, size_t ws_size,
                              hipStream_t stream) {
    const float*     x   = (const float*)d_in[0];          // [40000, 256]
    const long long* ei  = (const long long*)d_in[1];      // [2, 640000]
    const float*     W   = (const float*)d_in[2];          // [256, 128]
    const float*     b   = (const float*)d_in[3];          // [128]
    float*           out = (float*)d_out;                  // [40000, 128]

    const long long* src = ei;               // row 0
    const long long* dst = ei + N_EDGES;     // row 1

    // workspace layout: deg/dinv [40960 floats, aligned] | xw [40000*128 floats]
    float* deg = (float*)d_ws;
    float* xw  = deg + 40960;

    // 1) deg = 1
    k_init_deg<<<(N_NODES + 255) / 256, 256, 0, stream>>>(deg);
    // 2) deg += per-edge
    k_deg_atomic<<<(N_EDGES + 255) / 256, 256, 0, stream>>>(dst, deg);
    // 3) dinv in place
    k_dinv<<<(N_NODES + 255) / 256, 256, 0, stream>>>(deg);
    // 4) xw = x @ W  (2500 waves, 8 waves/block, W staged in LDS)
    {
        int waves = N_NODES / 16;                 // 2500
        int blocks = (waves * 32 + 255) / 256;    // 313
        k_gemm_wmma<<<blocks, 256, LDS_BYTES, stream>>>(x, W, xw);
    }
    // 5) out = dinv^2 * xw + b
    k_out_init<<<(N_NODES * OUT_DIM + 255) / 256, 256, 0, stream>>>(deg, xw, b, out);
    // 6) edge scatter (one wave/edge)
    {
        int blocks = (N_EDGES * 32) / 256;        // 80000
        k_scatter<<<blocks, 256, 0, stream>>>(src, dst, deg, xw, out);
    }
    // 7) relu
    k_relu<<<(N_NODES * OUT_DIM + 255) / 256, 256, 0, stream>>>(out);
}